// GateLoop_68272800137226
// MI455X (gfx1250) — compile-verified
//
#include <hip/hip_runtime.h>
#include <hip/hip_bf16.h>

// GateLoop: proj = x @ W + b ; inp=tanh, gates=sigmoid ; linear recurrence over S.
// ws layout: [0,64MB) tanh(inp) ; [64MB,128MB) sigmoid(input_gate) ;
//            then aggA (1MB), aggKV (1MB), carry (1MB). Requires ws_size >= ~132MB.
// d_out temporarily holds sigmoid(output_gate) between K1 and K4.

typedef float v2f __attribute__((ext_vector_type(2)));
typedef float v8f __attribute__((ext_vector_type(8)));

#define BATCH 4
#define SEQ   4096
#define DIN   1024
#define DH    1024
#define NOUT  (3 * DH)          // 3072
#define MTOT  (BATCH * SEQ)     // 16384

#define BLK_M 128
#define BLK_N 128
#define BLK_K 16
#define KPAIR (BLK_K / 2)       // 8 pair-rows for A (k even/odd packed in float2)
#define LDSS  144               // row stride (float2 for A, float for B): keeps halves on disjoint banks

#define NCHUNK 64
#define CLEN   (SEQ / NCHUNK)   // 64
#define BD     (BATCH * DH)     // 4096

__device__ __forceinline__ float fast_sigmoid(float x) {
    return __builtin_amdgcn_rcpf(1.0f + __expf(-x));
}
__device__ __forceinline__ float fast_tanh(float x) {
    // tanh(x) = 1 - 2/(1+e^{2x}); saturates correctly at +-inf
    return 1.0f - 2.0f * __builtin_amdgcn_rcpf(1.0f + __expf(2.0f * x));
}

// ---- CDNA5 async global->LDS copy (ASYNCcnt path) ----
__device__ __forceinline__ void async_cp_b128(const float* g, float* l) {
#if __has_builtin(__builtin_amdgcn_global_load_async_to_lds_b128)
    typedef int v4i_ __attribute__((vector_size(4 * sizeof(int))));
    typedef __attribute__((address_space(1))) v4i_* gp_t;
    typedef __attribute__((address_space(3))) v4i_* lp_t;
    __builtin_amdgcn_global_load_async_to_lds_b128(
        (gp_t)(void*)g, (lp_t)(void*)l, 0, 0);
#else
    unsigned lds_off =
        (unsigned)(unsigned long long)(__attribute__((address_space(3))) void*)l;
    unsigned long long ga = (unsigned long long)g;
    asm volatile("global_load_async_to_lds_b128 %0, %1, off"
                 :: "v"(lds_off), "v"(ga) : "memory");
#endif
}
__device__ __forceinline__ void wait_asynccnt0() {
#if __has_builtin(__builtin_amdgcn_s_wait_asynccnt)
    __builtin_amdgcn_s_wait_asynccnt(0);
#else
    asm volatile("s_wait_asynccnt 0x0" ::: "memory");
#endif
}

__global__ __launch_bounds__(256)
void gemm_act_kernel(const float* __restrict__ X, const float* __restrict__ W,
                     const float* __restrict__ bias,
                     float* __restrict__ ws_t, float* __restrict__ ws_g,
                     float* __restrict__ og_out)
{
    // A: [pair-row p][m] = (A[2p][m], A[2p+1][m]) as float2 (transposed in registers).
    // B: plain [k][n] float rows, filled by async global->LDS DMA (no transpose needed).
    __shared__ float2 Asp[2][KPAIR][LDSS];
    __shared__ float  BsF[2][BLK_K][LDSS];

    const int tid  = threadIdx.x;
    const int lane = tid & 31;
    const int wave = tid >> 5;       // 0..7
    const int wm   = wave & 3;       // 32-row band within 128
    const int wn   = wave >> 2;      // 64-col band within 128
    const int half = lane >> 4;      // 0/1: k-lo / k-hi half of the wave
    const int lidx = lane & 15;

    const int gm0 = blockIdx.y * BLK_M;
    const int gn0 = blockIdx.x * BLK_N;

    // A loader: rows (tid>>2)+{0,64}, k-quad tid&3
    const int ar  = tid >> 2;        // 0..63
    const int akq = tid & 3;         // k = 4*akq .. 4*akq+3
    // B async loader: row tid>>4 (0..15), two 16B chunks at n = 4*(tid&15) and +64
    const int bk  = tid >> 4;
    const int bn  = 4 * (tid & 15);

    v8f acc[2][4];
    const v8f vzero = {0.f, 0.f, 0.f, 0.f, 0.f, 0.f, 0.f, 0.f};
#pragma unroll
    for (int i = 0; i < 2; ++i)
#pragma unroll
        for (int j = 0; j < 4; ++j) acc[i][j] = vzero;

    float4 pa[2];

    auto load_a_global = [&](int k0) {
#pragma unroll
        for (int i = 0; i < 2; ++i)
            pa[i] = *(const float4*)(X + (gm0 + ar + 64 * i) * DIN + k0 + 4 * akq);
    };

    auto issue_b_async = [&](int k0, int buf) {
        const float* gb = W + (k0 + bk) * NOUT + gn0 + bn;
#pragma unroll
        for (int i = 0; i < 2; ++i)
            async_cp_b128(gb + 64 * i, &BsF[buf][bk][bn + 64 * i]);
    };

    auto store_a_lds = [&](int buf) {
        // Transpose 4 consecutive k of one row m into two pair-rows.
        // XOR swizzle on m (bits 3..4 by akq) -> conflict-free scatter stores.
        const int sw = akq << 3;
#pragma unroll
        for (int i = 0; i < 2; ++i) {
            const int m = (ar + 64 * i) ^ sw;
            Asp[buf][2 * akq + 0][m] = make_float2(pa[i].x, pa[i].y);
            Asp[buf][2 * akq + 1][m] = make_float2(pa[i].z, pa[i].w);
        }
    };

    auto load_frags = [&](int buf, int kk, v2f* af, v2f* bf) {
        const int p  = (kk >> 1) + half;     // A pair-row for this lane half
        const int sw = ((p >> 1) & 3) << 3;  // matches store-side swizzle
#pragma unroll
        for (int i = 0; i < 2; ++i) {
            const int m = (wm * 32 + i * 16 + lidx) ^ sw;
            const float2 v = Asp[buf][p][m];
            af[i][0] = v.x; af[i][1] = v.y;
        }
        const int p0 = kk + 2 * half;        // B k-row for this lane half
#pragma unroll
        for (int j = 0; j < 4; ++j) {
            const int n = wn * 64 + j * 16 + lidx;
            bf[j][0] = BsF[buf][p0 + 0][n];  // merges into ds_load_2addr_b32
            bf[j][1] = BsF[buf][p0 + 1][n];
        }
    };

    auto compute = [&](int buf) {
        v2f afr[2][2], bfr[2][4];
        load_frags(buf, 0, afr[0], bfr[0]);
#pragma unroll
        for (int kk = 0; kk < BLK_K; kk += 4) {
            const int cur = (kk >> 2) & 1;
            if (kk + 4 < BLK_K) load_frags(buf, kk + 4, afr[cur ^ 1], bfr[cur ^ 1]);
#pragma unroll
            for (int i = 0; i < 2; ++i)
#pragma unroll
                for (int j = 0; j < 4; ++j)
                    acc[i][j] = __builtin_amdgcn_wmma_f32_16x16x4_f32(
                        false, afr[cur][i], false, bfr[cur][j], (short)0,
                        acc[i][j], false, false);
        }
    };

    const int nk = DIN / BLK_K;  // 64
    issue_b_async(0, 0);
    load_a_global(0);
    store_a_lds(0);
    wait_asynccnt0();
    __syncthreads();
    for (int kt = 0; kt < nk; ++kt) {
        if (kt + 1 < nk) {
            issue_b_async((kt + 1) * BLK_K, (kt + 1) & 1);
            load_a_global((kt + 1) * BLK_K);
        }
        compute(kt & 1);
        if (kt + 1 < nk) store_a_lds((kt + 1) & 1);
        wait_asynccnt0();
        __syncthreads();
    }

    // Fused bias + activation epilogue. Block lies entirely in one 1024-col chunk.
    const int chunk = gn0 >> 10;  // 0: inp(tanh) 1: input_gate(sig) 2: output_gate(sig)
#pragma unroll
    for (int j = 0; j < 4; ++j) {
        const int gcol = gn0 + wn * 64 + j * 16 + lidx;
        const float bv = bias[gcol];
        const int hcol = gcol & (DH - 1);
#pragma unroll
        for (int i = 0; i < 2; ++i) {
            const int rbase = gm0 + wm * 32 + i * 16 + half * 8;
#pragma unroll
            for (int v = 0; v < 8; ++v) {
                const float p = acc[i][j][v] + bv;
                const size_t oidx = (size_t)(rbase + v) * DH + hcol;
                if (chunk == 0)      ws_t[oidx]   = fast_tanh(p);
                else if (chunk == 1) ws_g[oidx]   = fast_sigmoid(p);
                else                 og_out[oidx] = fast_sigmoid(p);
            }
        }
    }
}

// ---- chunked linear-recurrence scan: h[t] = (1-g[t])*h[t-1] + t[t]*g[t] ----

__global__ __launch_bounds__(256)
void chunk_reduce_kernel(const float* __restrict__ ws_t, const float* __restrict__ ws_g,
                         float* __restrict__ aggA, float* __restrict__ aggKV)
{
    const int gid = blockIdx.x * blockDim.x + threadIdx.x;  // B*DH*NCHUNK
    const int d = gid & (DH - 1);
    const int b = (gid >> 10) & (BATCH - 1);
    const int c = gid >> 12;
    size_t idx = (size_t)(b * SEQ + c * CLEN) * DH + d;
    float A = 1.0f, KV = 0.0f;
#pragma unroll 4
    for (int i = 0; i < CLEN; ++i, idx += DH) {
        const float t = ws_t[idx];
        const float g = ws_g[idx];
        const float a = 1.0f - g;
        A = a * A;
        KV = fmaf(a, KV, t * g);
    }
    aggA[gid]  = A;   // layout [c][b*DH+d] == gid
    aggKV[gid] = KV;
}

__global__ __launch_bounds__(256)
void chunk_scan_kernel(const float* __restrict__ aggA, const float* __restrict__ aggKV,
                       float* __restrict__ carry)
{
    const int bd = blockIdx.x * blockDim.x + threadIdx.x;  // 4096
    float run = 0.0f;
#pragma unroll 8
    for (int c = 0; c < NCHUNK; ++c) {
        const int idx = c * BD + bd;
        carry[idx] = run;                       // exclusive prefix (hidden entering chunk c)
        run = fmaf(aggA[idx], run, aggKV[idx]);
    }
}

__global__ __launch_bounds__(256)
void chunk_apply_kernel(const float* __restrict__ ws_t, const float* __restrict__ ws_g,
                        const float* __restrict__ carry, float* __restrict__ out)
{
    const int gid = blockIdx.x * blockDim.x + threadIdx.x;
    const int d = gid & (DH - 1);
    const int b = (gid >> 10) & (BATCH - 1);
    const int c = gid >> 12;
    size_t idx = (size_t)(b * SEQ + c * CLEN) * DH + d;
    float h = carry[gid];
#pragma unroll 4
    for (int i = 0; i < CLEN; ++i, idx += DH) {
        const float t = ws_t[idx];
        const float g = ws_g[idx];
        h = fmaf(1.0f - g, h, t * g);
        const float og = out[idx];              // sigmoid(output_gate) stashed by K1
        out[idx] = fast_tanh(h) * og;
    }
}

extern "C" void kernel_launch(void* const* d_in, const int* in_sizes, int n_in,
                              void* d_out, int out_size, void* d_ws, size_t ws_size,
                              hipStream_t stream)
{
    const float* x = (const float*)d_in[0];
    const float* W = (const float*)d_in[1];
    const float* b = (const float*)d_in[2];
    float* out = (float*)d_out;

    char* ws = (char*)d_ws;
    const size_t actBytes = (size_t)MTOT * DH * sizeof(float);     // 64 MB
    const size_t aggBytes = (size_t)BD * NCHUNK * sizeof(float);   // 1 MB
    float* ws_t  = (float*)(ws);
    float* ws_g  = (float*)(ws + actBytes);
    float* aggA  = (float*)(ws + 2 * actBytes);
    float* aggKV = (float*)(ws + 2 * actBytes + aggBytes);
    float* carry = (float*)(ws + 2 * actBytes + 2 * aggBytes);

    dim3 gg(NOUT / BLK_N, MTOT / BLK_M);  // (24, 128)
    gemm_act_kernel<<<gg, 256, 0, stream>>>(x, W, b, ws_t, ws_g, out);
    chunk_reduce_kernel<<<(BATCH * DH * NCHUNK) / 256, 256, 0, stream>>>(ws_t, ws_g, aggA, aggKV);
    chunk_scan_kernel<<<BD / 256, 256, 0, stream>>>(aggA, aggKV, carry);
    chunk_apply_kernel<<<(BATCH * DH * NCHUNK) / 256, 256, 0, stream>>>(ws_t, ws_g, carry, out);
}